// Ft_Surrogate_39307540693373
// MI455X (gfx1250) — compile-verified
//
#include <hip/hip_runtime.h>

typedef float v2f __attribute__((ext_vector_type(2)));
typedef float v8f __attribute__((ext_vector_type(8)));

#define PI2F 6.28318530717958647692f
constexpr int FFT_N = 8192;
constexpr float INV_N = 1.0f / 8192.0f;

// -------------------- WMMA helper: D(16x16 f32) = A(16x4) * B(4x16) + C ---
__device__ __forceinline__ v8f wmma4(v2f a, v2f b, v8f c) {
  // 8 args: (neg_a, A, neg_b, B, c_mod, C, reuse_a, reuse_b)
  return __builtin_amdgcn_wmma_f32_16x16x4_f32(false, a, false, b, (short)0, c,
                                               false, false);
}

// -------------------- radix-16 forward tile (DIF: matmul then twiddle) ----
// One wave handles a 16x16 tile: 16 columns at LDS addr0 + r*st (r = 0..15).
// D16[p][r] = cos(2pi p r/16) - i sin(...)  ->  Yre = C*Xre + S*Xim,
//                                               Yim = C*Xim + (-S)*Xre
__device__ __forceinline__ void fwd_tile(float* sRe, float* sIm, int addr0,
                                         int st, float twBase, const v2f* Ca,
                                         const v2f* Sa, const v2f* NSa,
                                         int half) {
  const int krow = 2 * half;
  v2f bre[4], bim[4];
#pragma unroll
  for (int kk = 0; kk < 4; ++kk) {
    int r0 = 4 * kk + krow;
    v2f tr, ti;
    tr.x = sRe[addr0 + r0 * st];
    tr.y = sRe[addr0 + (r0 + 1) * st];
    ti.x = sIm[addr0 + r0 * st];
    ti.y = sIm[addr0 + (r0 + 1) * st];
    bre[kk] = tr;
    bim[kk] = ti;
  }
  v8f yre = {0.f, 0.f, 0.f, 0.f, 0.f, 0.f, 0.f, 0.f};
  v8f yim = {0.f, 0.f, 0.f, 0.f, 0.f, 0.f, 0.f, 0.f};
#pragma unroll
  for (int kk = 0; kk < 4; ++kk) yre = wmma4(Ca[kk], bre[kk], yre);
#pragma unroll
  for (int kk = 0; kk < 4; ++kk) yre = wmma4(Sa[kk], bim[kk], yre);
#pragma unroll
  for (int kk = 0; kk < 4; ++kk) yim = wmma4(Ca[kk], bim[kk], yim);
#pragma unroll
  for (int kk = 0; kk < 4; ++kk) yim = wmma4(NSa[kk], bre[kk], yim);
// twiddle W_L^{p*s} = e^{-i*twBase*p} applied on output, then store in place
#pragma unroll
  for (int i = 0; i < 8; ++i) {
    int p = i + 8 * half;
    float ang = twBase * (float)p;
    float cs = __cosf(ang), sn = __sinf(ang);
    float re = yre[i] * cs + yim[i] * sn;
    float im = yim[i] * cs - yre[i] * sn;
    sRe[addr0 + p * st] = re;
    sIm[addr0 + p * st] = im;
  }
}

// -------------------- radix-16 inverse tile (DIT: twiddle then conj matmul)
// conj twiddle e^{+i*twBase*r} on load; conj(D16): Yre = C*Xre + (-S)*Xim,
//                                                  Yim = C*Xim + S*Xre
__device__ __forceinline__ void inv_tile(float* sRe, float* sIm, int addr0,
                                         int st, float twBase, const v2f* Ca,
                                         const v2f* Sa, const v2f* NSa,
                                         int half) {
  const int krow = 2 * half;
  v2f bre[4], bim[4];
#pragma unroll
  for (int kk = 0; kk < 4; ++kk) {
    int r0 = 4 * kk + krow;
    float a0 = twBase * (float)r0;
    float a1 = twBase * (float)(r0 + 1);
    float c0 = __cosf(a0), s0 = __sinf(a0);
    float c1 = __cosf(a1), s1 = __sinf(a1);
    float xr0 = sRe[addr0 + r0 * st], xi0 = sIm[addr0 + r0 * st];
    float xr1 = sRe[addr0 + (r0 + 1) * st], xi1 = sIm[addr0 + (r0 + 1) * st];
    v2f tr, ti;
    tr.x = xr0 * c0 - xi0 * s0;
    tr.y = xr1 * c1 - xi1 * s1;
    ti.x = xi0 * c0 + xr0 * s0;
    ti.y = xi1 * c1 + xr1 * s1;
    bre[kk] = tr;
    bim[kk] = ti;
  }
  v8f yre = {0.f, 0.f, 0.f, 0.f, 0.f, 0.f, 0.f, 0.f};
  v8f yim = {0.f, 0.f, 0.f, 0.f, 0.f, 0.f, 0.f, 0.f};
#pragma unroll
  for (int kk = 0; kk < 4; ++kk) yre = wmma4(Ca[kk], bre[kk], yre);
#pragma unroll
  for (int kk = 0; kk < 4; ++kk) yre = wmma4(NSa[kk], bim[kk], yre);
#pragma unroll
  for (int kk = 0; kk < 4; ++kk) yim = wmma4(Ca[kk], bim[kk], yim);
#pragma unroll
  for (int kk = 0; kk < 4; ++kk) yim = wmma4(Sa[kk], bre[kk], yim);
#pragma unroll
  for (int i = 0; i < 8; ++i) {
    int p = i + 8 * half;
    sRe[addr0 + p * st] = yre[i];
    sIm[addr0 + p * st] = yim[i];
  }
}

// -------------------- main per-signal kernel ------------------------------
// One block = one (b,c) signal, contiguous in sigIn ([B,C,T] layout).
// Forward FFT (three WMMA radix-16 stages + radix-2), amplitude + random
// phase, inverse network, write real part * 1/N in place.
__global__ __launch_bounds__(256) void fft_surrogate_kernel(
    const float* __restrict__ sigIn, const float* __restrict__ phT,
    const float* __restrict__ phRaw, float* __restrict__ sigOut, int usePhT) {
  __shared__ float sRe[FFT_N];
  __shared__ float sIm[FFT_N];

  const int tid = threadIdx.x;
  const int lane = tid & 31;
  const int wv = tid >> 5;
  const int half = lane >> 4;  // 0: lanes 0-15, 1: lanes 16-31
  const int l16 = lane & 15;
  const int sig = blockIdx.x;           // = b*64 + c
  const size_t base = (size_t)sig * FFT_N;

  // ---- load real signal (contiguous, float4), imag = 0 ----
  {
    const float4* in4 = (const float4*)(sigIn + base);
#pragma unroll
    for (int i = 0; i < 8; ++i) {
      int q = tid + i * 256;
      float4 v = in4[q];
      int e = q * 4;
      sRe[e + 0] = v.x;
      sRe[e + 1] = v.y;
      sRe[e + 2] = v.z;
      sRe[e + 3] = v.w;
      sIm[e + 0] = 0.f;
      sIm[e + 1] = 0.f;
      sIm[e + 2] = 0.f;
      sIm[e + 3] = 0.f;
    }
  }
  __syncthreads();

  // ---- DFT16 matrix in WMMA A-layout: lane l16 = row M=p,
  //      VGPR pair holds K = {4kk+2*half, +1} ----
  v2f Ca[4], Sa[4], NSa[4];
  {
    const float w16 = PI2F / 16.0f;
#pragma unroll
    for (int kk = 0; kk < 4; ++kk) {
      int r0 = 4 * kk + 2 * half;
      float a0 = w16 * (float)(l16 * r0);
      float a1 = w16 * (float)(l16 * (r0 + 1));
      v2f c, s, ns;
      c.x = cosf(a0);
      c.y = cosf(a1);
      s.x = sinf(a0);
      s.y = sinf(a1);
      ns.x = -s.x;
      ns.y = -s.y;
      Ca[kk] = c;
      Sa[kk] = s;
      NSa[kk] = ns;
    }
  }

  // ---- forward stage 1: radix-16, stride 512, twiddle W_8192^{p*s} ----
  for (int t = wv; t < 32; t += 8) {
    int col = t * 16 + l16;  // s in [0,512)
    fwd_tile(sRe, sIm, col, 512, (PI2F / 8192.0f) * (float)col, Ca, Sa, NSa,
             half);
  }
  __syncthreads();
  // ---- forward stage 2: radix-16, stride 32, twiddle W_512^{p2*s2} ----
  for (int t = wv; t < 32; t += 8) {
    int col = t * 16 + l16;
    int p = col >> 5, s2 = col & 31;
    fwd_tile(sRe, sIm, p * 512 + s2, 32, (PI2F / 512.0f) * (float)s2, Ca, Sa,
             NSa, half);
  }
  __syncthreads();
  // ---- forward stage 3: radix-16, stride 2, twiddle W_32^{p3*s3} ----
  for (int t = wv; t < 32; t += 8) {
    int col = t * 16 + l16;
    int blk = col >> 1, s3 = col & 1;
    fwd_tile(sRe, sIm, blk * 32 + s3, 2, (PI2F / 32.0f) * (float)s3, Ca, Sa,
             NSa, half);
  }
  __syncthreads();
  // ---- forward stage 4: radix-2, stride 1 ----
#pragma unroll
  for (int i = 0; i < 16; ++i) {
    int m = tid + i * 256;
    int e = 2 * m, o = 2 * m + 1;
    float ar = sRe[e], ai = sIm[e], br = sRe[o], bi = sIm[o];
    sRe[e] = ar + br;
    sIm[e] = ai + bi;
    sRe[o] = ar - br;
    sIm[o] = ai - bi;
  }
  __syncthreads();

  // ---- spectrum: amp * e^{i*2pi*phase[k]}, k = digit-reversed(j) ----
  {
    const int b = sig >> 6;
    const int cch = sig & 63;
#pragma unroll
    for (int i = 0; i < 32; ++i) {
      int j = tid + i * 256;
      float re = sRe[j], im = sIm[j];
      float amp = sqrtf(re * re + im * im);
      int k = (j >> 9) | (((j >> 5) & 15) << 4) | (((j >> 1) & 15) << 8) |
              ((j & 1) << 12);
      float pv = usePhT ? phT[base + k]
                        : phRaw[((size_t)b * FFT_N + k) * 64 + cch];
      float ph = pv * PI2F;
      sRe[j] = amp * cosf(ph);
      sIm[j] = amp * sinf(ph);
    }
  }
  __syncthreads();

  // ---- inverse network: reverse stage order, conj twiddles/matrix ----
  // inverse radix-2 (self-conjugate butterfly)
#pragma unroll
  for (int i = 0; i < 16; ++i) {
    int m = tid + i * 256;
    int e = 2 * m, o = 2 * m + 1;
    float ar = sRe[e], ai = sIm[e], br = sRe[o], bi = sIm[o];
    sRe[e] = ar + br;
    sIm[e] = ai + bi;
    sRe[o] = ar - br;
    sIm[o] = ai - bi;
  }
  __syncthreads();
  // inverse of stage 3
  for (int t = wv; t < 32; t += 8) {
    int col = t * 16 + l16;
    int blk = col >> 1, s3 = col & 1;
    inv_tile(sRe, sIm, blk * 32 + s3, 2, (PI2F / 32.0f) * (float)s3, Ca, Sa,
             NSa, half);
  }
  __syncthreads();
  // inverse of stage 2
  for (int t = wv; t < 32; t += 8) {
    int col = t * 16 + l16;
    int p = col >> 5, s2 = col & 31;
    inv_tile(sRe, sIm, p * 512 + s2, 32, (PI2F / 512.0f) * (float)s2, Ca, Sa,
             NSa, half);
  }
  __syncthreads();
  // inverse of stage 1
  for (int t = wv; t < 32; t += 8) {
    int col = t * 16 + l16;
    inv_tile(sRe, sIm, col, 512, (PI2F / 8192.0f) * (float)col, Ca, Sa, NSa,
             half);
  }
  __syncthreads();

  // ---- write real part * 1/N (in place over W1) ----
  {
    float4* out4 = (float4*)(sigOut + base);
#pragma unroll
    for (int i = 0; i < 8; ++i) {
      int q = tid + i * 256;
      int e = q * 4;
      float4 v;
      v.x = sRe[e + 0] * INV_N;
      v.y = sRe[e + 1] * INV_N;
      v.z = sRe[e + 2] * INV_N;
      v.w = sRe[e + 3] * INV_N;
      out4[q] = v;
    }
  }
}

// -------------------- [B,T,C] -> [B,C,T] transpose (2 arrays via z) -------
__global__ __launch_bounds__(512) void transpose_tc(
    const float* __restrict__ a, const float* __restrict__ b,
    float* __restrict__ wa, float* __restrict__ wb) {
  __shared__ float tile[64][65];
  const float* src = blockIdx.z ? b : a;
  float* dst = blockIdx.z ? wb : wa;
  const int bb = blockIdx.y;
  const int t0 = blockIdx.x * 64;
  const int tx = threadIdx.x, ty = threadIdx.y;
#pragma unroll
  for (int r = 0; r < 8; ++r) {
    int t = t0 + ty + r * 8;
    tile[ty + r * 8][tx] = src[((size_t)bb * 8192 + t) * 64 + tx];
  }
  __syncthreads();
#pragma unroll
  for (int r = 0; r < 8; ++r) {
    int c = ty + r * 8;
    dst[((size_t)bb * 64 + c) * 8192 + t0 + tx] = tile[tx][c];
  }
}

// -------------------- surrogate [B,C,T] -> out [B,T,C] with mask select ---
__global__ __launch_bounds__(512) void select_write(
    const float* __restrict__ surrT, const float* __restrict__ wav,
    const float* __restrict__ mask, float* __restrict__ out) {
  __shared__ float tile[64][65];
  const int bb = blockIdx.y;
  const int t0 = blockIdx.x * 64;
  const int tx = threadIdx.x, ty = threadIdx.y;
#pragma unroll
  for (int r = 0; r < 8; ++r) {
    int c = ty + r * 8;
    tile[tx][c] = surrT[((size_t)bb * 64 + c) * 8192 + t0 + tx];
  }
  __syncthreads();
  float m = mask[bb * 64 + tx];
  bool useS = m < 0.5f;
#pragma unroll
  for (int r = 0; r < 8; ++r) {
    int t = t0 + ty + r * 8;
    size_t idx = ((size_t)bb * 8192 + t) * 64 + tx;
    float w = wav[idx];
    out[idx] = useS ? tile[ty + r * 8][tx] : w;
  }
}

// -------------------- host launcher ---------------------------------------
extern "C" void kernel_launch(void* const* d_in, const int* in_sizes, int n_in,
                              void* d_out, int out_size, void* d_ws,
                              size_t ws_size, hipStream_t stream) {
  const float* wav = (const float*)d_in[0];   // [64, 8192, 64] f32
  const float* ph = (const float*)d_in[1];    // [64, 8192, 64] f32
  const float* mask = (const float*)d_in[2];  // [64, 64] f32
  float* out = (float*)d_out;

  const size_t SIG_BYTES = (size_t)64 * 64 * 8192 * sizeof(float);  // 134MB
  float* W1 = (float*)d_ws;                         // transposed waveforms,
                                                    // reused for surrogate
  float* W2 = (float*)((char*)d_ws + SIG_BYTES);    // transposed phases
  const int usePhT = (ws_size >= 2 * SIG_BYTES) ? 1 : 0;

  dim3 tb(64, 8);
  transpose_tc<<<dim3(128, 64, usePhT ? 2 : 1), tb, 0, stream>>>(wav, ph, W1,
                                                                 W2);
  fft_surrogate_kernel<<<4096, 256, 0, stream>>>(W1, W2, ph, W1, usePhT);
  select_write<<<dim3(128, 64), tb, 0, stream>>>(W1, wav, mask, out);
}